// S4_23914377904487
// MI455X (gfx1250) — compile-verified
//
#include <hip/hip_runtime.h>
#include <hip/hip_bf16.h>
#include <math.h>

typedef float v2f __attribute__((ext_vector_type(2)));
typedef float v8f __attribute__((ext_vector_type(8)));

#define B_DIM 4
#define H_DIM 256
#define L_DIM 4096
#define N2    32
#define TC    512
#define NC    8            // L_DIM / TC

__device__ __forceinline__ float wave_sum32(float v) {
#pragma unroll
  for (int m = 16; m >= 1; m >>= 1) v += __shfl_xor(v, m, 32);
  return v;
}

__device__ __forceinline__ float gelu_tanh(float y) {
  float y3 = y * y * y;
  return 0.5f * y * (1.0f + tanhf(0.7978845608028654f * (y + 0.044715f * y3)));
}

// ---------------------------------------------------------------------------
// Kernel 1: per-(h,n) discretization.  lane n of wave h computes
//   dA = (1+dt*w/2)/(1-dt*w/2),  B' = C * dt*B/(1-dt*w/2)   (complex)
// and per-h scalars  D[h], dCB[h] = 4 * sum(Re dB) * sum(Re C*dB).
// ---------------------------------------------------------------------------
__global__ void k_setup(const float* __restrict__ log_dt,
                        const float* __restrict__ w_re, const float* __restrict__ w_im,
                        const float* __restrict__ B_re, const float* __restrict__ B_im,
                        const float* __restrict__ C_re, const float* __restrict__ C_im,
                        const float* __restrict__ D,
                        float4* __restrict__ params, float2* __restrict__ hcoef) {
  int h = blockIdx.x;
  int n = threadIdx.x;            // 0..31
  int idx = h * N2 + n;

  float dt = expf(log_dt[h]);
  float zr = 0.5f * dt * w_re[idx];
  float zi = 0.5f * dt * w_im[idx];
  float dr = 1.0f - zr, di = -zi;
  float inv = 1.0f / (dr * dr + di * di);
  float nr = 1.0f + zr, ni = zi;
  float Ar = (nr * dr + ni * di) * inv;
  float Ai = (ni * dr - nr * di) * inv;

  float br = B_re[idx], bi = B_im[idx];
  float dBr = dt * (br * dr + bi * di) * inv;
  float dBi = dt * (bi * dr - br * di) * inv;

  float Cr = C_re[idx], Ci = C_im[idx];          // C==1 channel
  float Bpr = Cr * dBr - Ci * dBi;               // Re(C*dB)
  float Bpi = Cr * dBi + Ci * dBr;

  float sdB = wave_sum32(dBr);
  float sCB = wave_sum32(Bpr);

  params[idx] = make_float4(Ar, Ai, Bpr, Bpi);
  if (n == 0) hcoef[h] = make_float2(D[h], 4.0f * sdB * sCB);
}

// ---------------------------------------------------------------------------
// Kernel 2: chunk scan from zero state; emit only chunk-end state.
// One wave per (b,h,chunk); lane = state n.  u broadcast via v_readlane.
// ---------------------------------------------------------------------------
__global__ void k_scan1(const float* __restrict__ u,
                        const float4* __restrict__ params,
                        float2* __restrict__ zend) {
  int wid  = blockIdx.x * (blockDim.x >> 5) + (threadIdx.x >> 5);
  int lane = threadIdx.x & 31;
  int c = wid & (NC - 1);
  int h = (wid / NC) & (H_DIM - 1);
  int b = wid / (NC * H_DIM);

  float4 p = params[h * N2 + lane];
  const float* urow = u + (size_t)(b * H_DIM + h) * L_DIM + c * TC;

  float zr = 0.0f, zi = 0.0f;
  for (int t0 = 0; t0 < TC; t0 += 32) {
    float uv = urow[t0 + lane];
#pragma unroll
    for (int t = 0; t < 32; ++t) {
      float us = __int_as_float(__builtin_amdgcn_readlane(__float_as_int(uv), t));
      float nzr = fmaf(p.x, zr, fmaf(-p.y, zi, p.z * us));
      float nzi = fmaf(p.x, zi, fmaf( p.y, zr, p.w * us));
      zr = nzr; zi = nzi;
    }
  }
  zend[(size_t)wid * N2 + lane] = make_float2(zr, zi);
}

// ---------------------------------------------------------------------------
// Kernel 3: propagate chunk-initial states: zinit[c] = sum_{j<c} dA^{(c-j)Tc} zend[j]
// via carry = dA^Tc * (carry + zend[c]).  dA^512 by 9 complex squarings.
// ---------------------------------------------------------------------------
__global__ void k_fix(const float4* __restrict__ params,
                      const float2* __restrict__ zend,
                      float2* __restrict__ zinit) {
  int wid  = blockIdx.x * (blockDim.x >> 5) + (threadIdx.x >> 5);   // b*H + h
  int lane = threadIdx.x & 31;
  int h = wid & (H_DIM - 1);

  float4 p = params[h * N2 + lane];
  float ar = p.x, ai = p.y;
#pragma unroll
  for (int s = 0; s < 9; ++s) {            // dA^(2^9) = dA^512
    float tr = ar * ar - ai * ai;
    ai = 2.0f * ar * ai;
    ar = tr;
  }
  float cr = 0.0f, ci = 0.0f;
#pragma unroll
  for (int c = 0; c < NC; ++c) {
    size_t idx = ((size_t)wid * NC + c) * N2 + lane;
    zinit[idx] = make_float2(cr, ci);
    float2 ze = zend[idx];
    float tr = cr + ze.x, ti = ci + ze.y;
    cr = ar * tr - ai * ti;
    ci = ar * ti + ai * tr;
  }
}

// ---------------------------------------------------------------------------
// Kernel 4: chunk scan from true initial state; per 32-step tile do an LDS
// transpose-reduce over the 32 states, add D-skip + liquid term, gelu, write g.
// One wave (32-thread block) per (b,h,chunk).
// ---------------------------------------------------------------------------
__global__ void k_scan2(const float* __restrict__ u,
                        const float4* __restrict__ params,
                        const float2* __restrict__ hcoef,
                        const float2* __restrict__ zinit,
                        float* __restrict__ g) {
  __shared__ float red[32 * 34];           // stride 34 words: 8B aligned rows, conflict-free
  int wid  = blockIdx.x;
  int lane = threadIdx.x;
  int c = wid & (NC - 1);
  int h = (wid / NC) & (H_DIM - 1);
  int b = wid / (NC * H_DIM);

  float4 p  = params[h * N2 + lane];
  float2 hc = hcoef[h];                    // (D[h], dCB[h])
  float2 z0 = zinit[(size_t)wid * N2 + lane];
  float zr = z0.x, zi = z0.y;

  const float* ur = u + (size_t)(b * H_DIM + h) * L_DIM;
  float* grow     = g + (size_t)(b * H_DIM + h) * L_DIM;
  int lbase = c * TC;

  for (int t0 = 0; t0 < TC; t0 += 32) {
    float uv = ur[lbase + t0 + lane];
    int   gl = lbase + t0 + lane - 1;
    float up = (gl >= 0) ? ur[gl] : 0.0f;

    float cbuf[32];
#pragma unroll
    for (int t = 0; t < 32; ++t) {
      float us = __int_as_float(__builtin_amdgcn_readlane(__float_as_int(uv), t));
      float nzr = fmaf(p.x, zr, fmaf(-p.y, zi, p.z * us));
      float nzi = fmaf(p.x, zi, fmaf( p.y, zr, p.w * us));
      zr = nzr; zi = nzi;
      cbuf[t] = zr;                        // Re(z_n) contribution at step t
    }

    __syncthreads();
#pragma unroll
    for (int t = 0; t < 32; ++t) red[t * 34 + lane] = cbuf[t];
    __syncthreads();

    float s = 0.0f;
    const float2* row = (const float2*)(red + lane * 34);
#pragma unroll
    for (int k2 = 0; k2 < 16; ++k2) { float2 v = row[k2]; s += v.x + v.y; }

    // lane t owns l = lbase+t0+t; its own uv is exactly u_l, up is u_{l-1}
    float y = 2.0f * s + hc.x * uv + hc.y * (uv * up);
    grow[lbase + t0 + lane] = gelu_tanh(y);
  }
}

// ---------------------------------------------------------------------------
// Kernel 5: out[b,o,l] = sum_h W[o,h] * g[b,h,l] + bias[o]
// One wave per 16x16 output tile, K-loop of v_wmma_f32_16x16x4_f32.
// A = W tile (16x4), B = g tile (4x16), f32 accumulation.
// ---------------------------------------------------------------------------
__global__ void k_gemm(const float* __restrict__ g,
                       const float* __restrict__ W,
                       const float* __restrict__ bias,
                       float* __restrict__ out) {
  int wid  = blockIdx.x * (blockDim.x >> 5) + (threadIdx.x >> 5);  // 0..16383
  int lane = threadIdx.x & 31;
  int lt = wid & 255;              // l tile (16 cols)
  int ot = (wid >> 8) & 15;        // o tile (16 rows)
  int b  = wid >> 12;

  int n  = lane & 15;              // N index (l)  / M row for A
  int hh = lane >> 4;              // half-wave select

  const float* Wrow = W + (size_t)(ot * 16 + n) * H_DIM;   // A: M = lane&15
  const float* gcol = g + (size_t)b * H_DIM * L_DIM + lt * 16 + n;

  v8f acc = {};
  for (int k0 = 0; k0 < H_DIM; k0 += 4) {
    int ka = k0 + hh * 2;                          // K pair held by this half-wave
    v2f a = *(const v2f*)(Wrow + ka);              // A[m][ka], A[m][ka+1]
    v2f bm;
    bm.x = gcol[(size_t)ka * L_DIM];               // B[ka][n]
    bm.y = gcol[(size_t)(ka + 1) * L_DIM];         // B[ka+1][n]
    acc = __builtin_amdgcn_wmma_f32_16x16x4_f32(
        false, a, false, bm, (short)0, acc, false, false);
  }

#pragma unroll
  for (int j = 0; j < 8; ++j) {
    int o = ot * 16 + j + hh * 8;                  // C/D layout: M = j + 8*(lane>=16)
    out[((size_t)(b * H_DIM) + o) * L_DIM + lt * 16 + n] = acc[j] + bias[o];
  }
}

// ---------------------------------------------------------------------------
extern "C" void kernel_launch(void* const* d_in, const int* in_sizes, int n_in,
                              void* d_out, int out_size, void* d_ws, size_t ws_size,
                              hipStream_t stream) {
  const float* u      = (const float*)d_in[0];
  const float* log_dt = (const float*)d_in[1];
  const float* w_re   = (const float*)d_in[2];
  const float* w_im   = (const float*)d_in[3];
  const float* B_re   = (const float*)d_in[4];
  const float* B_im   = (const float*)d_in[5];
  const float* C_re   = (const float*)d_in[6];
  const float* C_im   = (const float*)d_in[7];
  const float* D      = (const float*)d_in[8];
  const float* W      = (const float*)d_in[9];
  const float* bias   = (const float*)d_in[10];
  float* out = (float*)d_out;

  // workspace layout (floats)
  float* ws = (float*)d_ws;
  float4* params = (float4*)ws;                             // H*32*4      = 32768 f
  float2* hcoef  = (float2*)(ws + 32768);                   // H*2         =   512 f
  float2* zend   = (float2*)(ws + 33280);                   // B*H*NC*32*2 = 524288 f
  float2* zinit  = (float2*)(ws + 33280 + 524288);          // 524288 f
  float*  g      = ws + 33280 + 2 * 524288;                 // B*H*L       = 4194304 f

  k_setup<<<H_DIM, 32, 0, stream>>>(log_dt, w_re, w_im, B_re, B_im,
                                    C_re, C_im, D, params, hcoef);

  // B*H*NC = 8192 waves, 8 waves / 256-thread block
  k_scan1<<<(B_DIM * H_DIM * NC) / 8, 256, 0, stream>>>(u, params, zend);

  // B*H = 1024 waves
  k_fix<<<(B_DIM * H_DIM) / 8, 256, 0, stream>>>(params, zend, zinit);

  // one wave per (b,h,chunk), needs private LDS transpose buffer
  k_scan2<<<B_DIM * H_DIM * NC, 32, 0, stream>>>(u, params, hcoef, zinit, g);

  // 4 * 16 * 256 = 16384 output tiles, 8 waves / block
  k_gemm<<<(B_DIM * 16 * 256) / 8, 256, 0, stream>>>(g, W, bias, out);
}